// StandardROIHeads_70274254897560
// MI455X (gfx1250) — compile-verified
//
#include <hip/hip_runtime.h>
#include <hip/hip_bf16.h>

typedef __attribute__((ext_vector_type(2)))  float  v2f;
typedef __attribute__((ext_vector_type(8)))  float  v8f;

#define BATCH 512
#define NPOS  128
#define MAXM  256
#define SORTN 8192
#define WAVES 8          // waves per match block (16 proposals each)

// ---------- deterministic hash (replaces jax.random for sampling scores) ----------
__device__ __forceinline__ unsigned hash3(unsigned a, unsigned b, unsigned c) {
  unsigned h = a * 0x9E3779B1u + b * 0x85EBCA77u + c * 0xC2B2AE3Du + 0x27D4EB2Fu;
  h ^= h >> 16; h *= 0x7FEB352Du;
  h ^= h >> 15; h *= 0x846CA68Bu;
  h ^= h >> 16;
  return h;
}

// fast reciprocal: v_rcp_f32 + one Newton-Raphson step (~0.5 ulp).
// Denominator is a strictly positive box-area sum, so no special cases.
__device__ __forceinline__ float fast_rcp(float d) {
  float r = __builtin_amdgcn_rcpf(d);
  r = r + r * __builtin_fmaf(-d, r, 1.0f);
  return r;
}

// D[m][n] = area_g[m] + area_p[n] for a 16x16 tile via rank-2 f32 WMMA.
// hlf==0 lanes carry K=0,1 (live); hlf==1 lanes carry K=2,3 (zero).
__device__ __forceinline__ v8f denom_tile(float ag, float ap, int hlf) {
  v8f c = {0.f, 0.f, 0.f, 0.f, 0.f, 0.f, 0.f, 0.f};
  v2f a, b;
  a.x = hlf ? 0.0f : ag;    // A[m][K=0] = area_g[m]
  a.y = hlf ? 0.0f : 1.0f;  // A[m][K=1] = 1
  b.x = hlf ? 0.0f : 1.0f;  // B[K=0][n] = 1
  b.y = hlf ? 0.0f : ap;    // B[K=1][n] = area_p[n]
  return __builtin_amdgcn_wmma_f32_16x16x4_f32(false, a, false, b, (short)0, c,
                                               false, false);
}

// ------------- kernel 1: IoU matching (8 waves/block, 16 proposals/wave) -------------
__global__ __launch_bounds__(32 * WAVES)
void match_kernel(const float* __restrict__ gt_boxes,   // [N,M,4]
                  const int*   __restrict__ gt_classes, // [N,M]
                  const float* __restrict__ pr_boxes,   // [N,R,4]
                  int* __restrict__ matched_idx,        // [N,R]
                  int* __restrict__ labels,             // [N,R]
                  int M, int R) {
  __shared__ float s_gt[MAXM * 4];
  __shared__ float s_area[MAXM];
  const int img  = blockIdx.y;
  const int lane = threadIdx.x;            // 0..31 (wave32)
  const int wv   = threadIdx.y;            // 0..WAVES-1
  const int col  = lane & 15;
  const int hlf  = lane >> 4;              // 0: C rows 0..7 ; 1: rows 8..15
  const int Mc   = (M < MAXM) ? M : MAXM;
  const int n0   = (blockIdx.x * WAVES + wv) * 16;

  // stage GT boxes + areas once per 128 proposals (whole block cooperates)
  const float* gtb = gt_boxes + (long)img * M * 4;
  for (int i = wv * 32 + lane; i < Mc; i += 32 * WAVES) {
    float x1 = gtb[i * 4 + 0], y1 = gtb[i * 4 + 1];
    float x2 = gtb[i * 4 + 2], y2 = gtb[i * 4 + 3];
    s_gt[i * 4 + 0] = x1; s_gt[i * 4 + 1] = y1;
    s_gt[i * 4 + 2] = x2; s_gt[i * 4 + 3] = y2;
    s_area[i] = (x2 - x1) * (y2 - y1);
  }
  __syncthreads();

  int n = n0 + col;
  if (n >= R) n = R - 1;                   // clamp; final store is guarded
  const float* pb = pr_boxes + ((long)img * R + n) * 4;
  float px1 = pb[0], py1 = pb[1], px2 = pb[2], py2 = pb[3];
  float ap = (px2 - px1) * (py2 - py1);

  float maxv = -1.0f;
  int   maxi = 0;
  const int ntile = (Mc + 15) >> 4;
  for (int t = 0; t < ntile; ++t) {
    int g0 = t * 16;
    int gm = g0 + col; if (gm >= Mc) gm = Mc - 1;
    v8f dsum = denom_tile(s_area[gm], ap, hlf);   // full-EXEC WMMA
#pragma unroll
    for (int i = 0; i < 8; ++i) {
      int m  = g0 + hlf * 8 + i;
      int mc = (m < Mc) ? m : (Mc - 1);           // clamped LDS index (branchless)
      float gx1 = s_gt[mc * 4 + 0], gy1 = s_gt[mc * 4 + 1];
      float gx2 = s_gt[mc * 4 + 2], gy2 = s_gt[mc * 4 + 3];
      float w = fmaxf(fminf(gx2, px2) - fmaxf(gx1, px1), 0.0f);
      float h = fmaxf(fminf(gy2, py2) - fmaxf(gy1, py1), 0.0f);
      float inter = w * h;
      float iou = inter * fast_rcp(dsum[i] - inter); // denom = ag+ap-inter
      bool upd = (m < Mc) && (iou > maxv);           // first-occurrence argmax
      maxv = upd ? iou : maxv;
      maxi = upd ? m   : maxi;
    }
  }

  // merge row-halves within the wave (wave32 shuffle); tie-break on smaller idx
  float ov = __shfl_xor(maxv, 16, 32);
  int   oi = __shfl_xor(maxi, 16, 32);
  if (ov > maxv || (ov == maxv && oi < maxi)) { maxv = ov; maxi = oi; }

  if (hlf == 0 && (n0 + col) < R) {
    int cls = gt_classes[(long)img * M + maxi];
    int lab = (maxv >= 0.5f) ? cls : 0;    // FG_IOU==BG_IOU -> no ignore band
    matched_idx[(long)img * R + n] = maxi;
    labels[(long)img * R + n]      = lab;
  }
}

// ------------- kernel 2: per-image sample + gather (1 block = 1 image) -------------
__global__ __launch_bounds__(1024)
void sample_kernel(const float* __restrict__ gt_boxes,   // [N,M,4]
                   const float* __restrict__ pr_boxes,   // [N,R,4]
                   const int*   __restrict__ matched_idx,// [N,R]
                   const int*   __restrict__ labels,     // [N,R]
                   float* __restrict__ out_boxes,        // [N,512,8]
                   float* __restrict__ out_cls,          // [N,512]
                   int M, int R) {
  __shared__ unsigned s_key[SORTN];
  __shared__ int s_fg[NPOS];
  __shared__ int s_bg[BATCH];
  __shared__ int s_cnt[2];
  const int img = blockIdx.x;
  const int tid = threadIdx.x;
  const int nth = blockDim.x;
  const int* L  = labels + (long)img * R;

  if (tid < 2) s_cnt[tid] = 0;
  __syncthreads();

  for (int pass = 0; pass < 2; ++pass) {
    // key = (18-bit hash score << 13) | index  for pool members (< 0x80000000)
    //       0x80000000|index for non-members, 0xFFFFFFFF for padding
    for (int r = tid; r < SORTN; r += nth) {
      unsigned k = 0xFFFFFFFFu;
      if (r < R) {
        int l = L[r];
        bool sel = (pass == 0) ? (l > 0) : (l == 0);
        if (sel) {
          unsigned h = hash3((unsigned)img, (unsigned)r, pass ? 0xBEEFu : 0xF00Du);
          k = ((h >> 14) << 13) | (unsigned)r;
          atomicAdd(&s_cnt[pass], 1);
        } else {
          k = 0x80000000u | (unsigned)r;
        }
      }
      s_key[r] = k;
    }
    // in-LDS bitonic sort, ascending (index bits give stable tie-break)
    for (int kk = 2; kk <= SORTN; kk <<= 1) {
      for (int j = kk >> 1; j > 0; j >>= 1) {
        __syncthreads();
        for (int i = tid; i < SORTN; i += nth) {
          int ixj = i ^ j;
          if (ixj > i) {
            unsigned a = s_key[i], b = s_key[ixj];
            bool up = ((i & kk) == 0);
            if ((a > b) == up) { s_key[i] = b; s_key[ixj] = a; }
          }
        }
      }
    }
    __syncthreads();
    if (pass == 0) {
      for (int i = tid; i < NPOS; i += nth) s_fg[i] = (int)(s_key[i] & 0x1FFFu);
    } else {
      for (int i = tid; i < BATCH; i += nth) s_bg[i] = (int)(s_key[i] & 0x1FFFu);
    }
    __syncthreads();
  }

  int num_pos = min(s_cnt[0], NPOS);
  int num_neg = min(s_cnt[1], BATCH - num_pos);

  for (int s = tid; s < BATCH; s += nth) {
    int sampled;
    if (s < num_pos) {
      sampled = s_fg[s];
    } else {
      int t = s - num_pos;
      if (t > BATCH - 1) t = BATCH - 1;
      sampled = s_bg[t];
    }
    bool valid = s < (num_pos + num_neg);
    if (sampled >= R) { sampled = 0; valid = false; }  // padding keys (never valid)
    int m = matched_idx[(long)img * R + sampled];
    if (m < 0) m = 0; if (m >= M) m = M - 1;
    int l = L[sampled];
    float v = valid ? 1.0f : 0.0f;
    const float* pb = pr_boxes + ((long)img * R + sampled) * 4;
    const float* gb = gt_boxes + ((long)img * M + m) * 4;
    float* ob = out_boxes + ((long)img * BATCH + s) * 8;
    ob[0] = pb[0] * v; ob[1] = pb[1] * v; ob[2] = pb[2] * v; ob[3] = pb[3] * v;
    ob[4] = gb[0] * v; ob[5] = gb[1] * v; ob[6] = gb[2] * v; ob[7] = gb[3] * v;
    out_cls[(long)img * BATCH + s] = valid ? (float)l : -1.0f;
  }
}

extern "C" void kernel_launch(void* const* d_in, const int* in_sizes, int n_in,
                              void* d_out, int out_size, void* d_ws, size_t ws_size,
                              hipStream_t stream) {
  (void)n_in; (void)ws_size;
  const float* gt_boxes   = (const float*)d_in[0];  // [N,M,4] f32
  const int*   gt_classes = (const int*)  d_in[1];  // [N,M]   i32
  const float* pr_boxes   = (const float*)d_in[2];  // [N,R,4] f32

  // per image: 512*8 box floats + 512 class values = 4608 output elements
  int N = out_size / (BATCH * 9);
  if (N <= 0) N = 1;
  int M = in_sizes[1] / N;
  int R = in_sizes[2] / (4 * N);

  int* matched_idx = (int*)d_ws;                      // [N,R]
  int* labels      = matched_idx + (long)N * R;       // [N,R]
  float* out_boxes = (float*)d_out;                   // [N,512,8]
  float* out_cls   = out_boxes + (long)N * BATCH * 8; // [N,512]

  dim3 g1((R + 16 * WAVES - 1) / (16 * WAVES), N);
  dim3 b1(32, WAVES);
  match_kernel<<<g1, b1, 0, stream>>>(gt_boxes, gt_classes, pr_boxes,
                                      matched_idx, labels, M, R);
  sample_kernel<<<N, 1024, 0, stream>>>(gt_boxes, pr_boxes, matched_idx, labels,
                                        out_boxes, out_cls, M, R);
}